// SpanEncoder_1494648619662
// MI455X (gfx1250) — compile-verified
//
#include <hip/hip_runtime.h>

typedef __attribute__((ext_vector_type(2))) float v2f;
typedef __attribute__((ext_vector_type(4))) float v4f;
typedef __attribute__((ext_vector_type(8))) float v8f;

#define S_LEN 4096
#define D_DIM 1024
#define K_DIM 2048   // 2*D
#define NROWS 2048   // B*M output rows

// ---------------------------------------------------------------------------
// Pack W[n][k] (row-major [1024, 2048]) into Wp[k/4][n][k%4] so that the two
// B-fragment elements each lane needs (k+{0,1} or k+{2,3} at fixed n) are an
// adjacent float2, and lanes 0..15 sweep contiguous n. 8 MB, written to d_ws.
// ---------------------------------------------------------------------------
__global__ __launch_bounds__(256)
void pack_w_kernel(const float* __restrict__ W, float* __restrict__ Wp) {
    int t  = blockIdx.x * blockDim.x + threadIdx.x;   // 0 .. 524287
    int n  = t & (D_DIM - 1);                         // 0..1023
    int kq = t >> 10;                                 // 0..511 (k = 4*kq)
    v4f v = *(const v4f*)(W + (size_t)n * K_DIM + (kq << 2));
    *(v4f*)(Wp + (size_t)kq * (D_DIM * 4) + n * 4) = v;  // coalesced store
}

// ---------------------------------------------------------------------------
// Gather + GEMM: out[m][n] = sum_k A[m][k] * W[n][k] + b[n]
//   A[m][k] = text[batch(m)][start(m)][k]        for k < 1024
//           = text[batch(m)][end(m)][k-1024]     for k >= 1024
// One wave -> 16 rows x 64 cols (4 x v_wmma_f32_16x16x4_f32 accumulators).
// Block = 8 waves -> 16 x 512 tile. Grid = (128 row tiles, 2 col halves).
// K-loop split into start-row half / end-row half so every A load is a plain
// base+immediate global_load_b64 (no per-iteration 64-bit address math).
// ---------------------------------------------------------------------------
__global__ __launch_bounds__(256)
void span_gemm_kernel(const float* __restrict__ text,
                      const int*   __restrict__ offs,
                      const int*   __restrict__ lens,
                      const float* __restrict__ Wp,
                      const float* __restrict__ bias,
                      float*       __restrict__ out) {
    const int lane = threadIdx.x & 31;
    const int wave = threadIdx.x >> 5;                 // 0..7
    const int rt   = blockIdx.x;                       // 0..127 row tile
    const int n0   = blockIdx.y * 512 + wave * 64;     // wave's col base

    const int mrow = lane & 15;
    const int koff = (lane < 16) ? 0 : 2;              // A/B K sub-offset
    const int m    = rt * 16 + mrow;                   // global mention row
    const int batch = m >> 6;                          // 64 mentions / batch

    const int s_idx = offs[m];
    const int e_idx = s_idx + lens[m] - 1;
    const float* srow = text + ((size_t)batch * S_LEN + s_idx) * D_DIM + koff;
    const float* erow = text + ((size_t)batch * S_LEN + e_idx) * D_DIM + koff;

    // Per-lane base into packed W: column n0+mrow, sub-k koff.
    const float* wp0 = Wp + (size_t)(n0 + mrow) * 4 + koff;             // k in [0,1024)
    const float* wp1 = wp0 + (size_t)256 * (D_DIM * 4);                  // k in [1024,2048)

    v8f acc0 = {}, acc1 = {}, acc2 = {}, acc3 = {};

    // ---- first half: start-token rows, k = 0..1023 ----
    #pragma unroll 4
    for (int kq = 0; kq < 256; ++kq) {
        v2f a  = *(const v2f*)(srow + (kq << 2));
        const float* wrow = wp0 + (size_t)kq * (D_DIM * 4);
        v2f b0 = *(const v2f*)(wrow + 0 * 64);
        v2f b1 = *(const v2f*)(wrow + 1 * 64);
        v2f b2 = *(const v2f*)(wrow + 2 * 64);
        v2f b3 = *(const v2f*)(wrow + 3 * 64);
        acc0 = __builtin_amdgcn_wmma_f32_16x16x4_f32(false, a, false, b0,
                                                     (short)0, acc0, false, false);
        acc1 = __builtin_amdgcn_wmma_f32_16x16x4_f32(false, a, false, b1,
                                                     (short)0, acc1, false, false);
        acc2 = __builtin_amdgcn_wmma_f32_16x16x4_f32(false, a, false, b2,
                                                     (short)0, acc2, false, false);
        acc3 = __builtin_amdgcn_wmma_f32_16x16x4_f32(false, a, false, b3,
                                                     (short)0, acc3, false, false);
    }

    // ---- second half: end-token rows, k = 1024..2047 ----
    #pragma unroll 4
    for (int kq = 0; kq < 256; ++kq) {
        v2f a  = *(const v2f*)(erow + (kq << 2));
        const float* wrow = wp1 + (size_t)kq * (D_DIM * 4);
        v2f b0 = *(const v2f*)(wrow + 0 * 64);
        v2f b1 = *(const v2f*)(wrow + 1 * 64);
        v2f b2 = *(const v2f*)(wrow + 2 * 64);
        v2f b3 = *(const v2f*)(wrow + 3 * 64);
        acc0 = __builtin_amdgcn_wmma_f32_16x16x4_f32(false, a, false, b0,
                                                     (short)0, acc0, false, false);
        acc1 = __builtin_amdgcn_wmma_f32_16x16x4_f32(false, a, false, b1,
                                                     (short)0, acc1, false, false);
        acc2 = __builtin_amdgcn_wmma_f32_16x16x4_f32(false, a, false, b2,
                                                     (short)0, acc2, false, false);
        acc3 = __builtin_amdgcn_wmma_f32_16x16x4_f32(false, a, false, b3,
                                                     (short)0, acc3, false, false);
    }

    // C/D layout: VGPR r -> row (r + 8*(lane>=16)), col n0 + (lane&15).
    const int rbase = rt * 16 + ((lane < 16) ? 0 : 8);

    {
        const int n = n0 + 0 * 16 + mrow;
        const float bv = bias[n];
        #pragma unroll
        for (int r = 0; r < 8; ++r)
            out[(size_t)(rbase + r) * D_DIM + n] = acc0[r] + bv;
    }
    {
        const int n = n0 + 1 * 16 + mrow;
        const float bv = bias[n];
        #pragma unroll
        for (int r = 0; r < 8; ++r)
            out[(size_t)(rbase + r) * D_DIM + n] = acc1[r] + bv;
    }
    {
        const int n = n0 + 2 * 16 + mrow;
        const float bv = bias[n];
        #pragma unroll
        for (int r = 0; r < 8; ++r)
            out[(size_t)(rbase + r) * D_DIM + n] = acc2[r] + bv;
    }
    {
        const int n = n0 + 3 * 16 + mrow;
        const float bv = bias[n];
        #pragma unroll
        for (int r = 0; r < 8; ++r)
            out[(size_t)(rbase + r) * D_DIM + n] = acc3[r] + bv;
    }
}

extern "C" void kernel_launch(void* const* d_in, const int* in_sizes, int n_in,
                              void* d_out, int out_size, void* d_ws, size_t ws_size,
                              hipStream_t stream) {
    const float* text = (const float*)d_in[0];   // [32, 4096, 1024] f32
    const int*   offs = (const int*)  d_in[1];   // [32, 64] i32
    const int*   lens = (const int*)  d_in[2];   // [32, 64] i32
    const float* W    = (const float*)d_in[3];   // [1024, 2048] f32
    const float* bias = (const float*)d_in[4];   // [1024] f32
    float*       out  = (float*)d_out;           // [32, 64, 1024] f32
    float*       Wp   = (float*)d_ws;            // needs 8 MB scratch

    // One-shot repack of W into WMMA-B-friendly layout (overwrites ws fully
    // every call -> deterministic).
    pack_w_kernel<<<(D_DIM * 512) / 256, 256, 0, stream>>>(W, Wp);

    dim3 grid(NROWS / 16, 2);   // 128 row tiles x 2 column halves
    span_gemm_kernel<<<grid, 256, 0, stream>>>(text, offs, lens, Wp, bias, out);
}